// MambaFeatureEnhancer_4303557230700
// MI455X (gfx1250) — compile-verified
//
#include <hip/hip_runtime.h>
#include <hip/hip_bf16.h>
#include <math.h>

typedef __attribute__((ext_vector_type(2))) float v2f;
typedef __attribute__((ext_vector_type(8))) float v8f;

namespace {

constexpr int Bn = 2, Cn = 128, Hn = 64, Wn = 64, Ln = Hn * Wn;
constexpr int DIn = 256, Sn = 16, Rn = 8, DKn = 4, NLn = 2, FFn = 512;
constexpr int BL = Bn * Ln;                  // 8192 tokens
constexpr int XPn = Rn + 2 * Sn;             // 40

constexpr int GF_BIAS = 1, GF_GELU = 2, GF_SOFTPLUS = 4, GF_RES = 8, GF_ACC = 16;

__device__ __forceinline__ float gelu_exact(float x) {
  return 0.5f * x * (1.0f + erff(x * 0.70710678118654752440f));
}
__device__ __forceinline__ float softplus_f(float x) {
  return fmaxf(x, 0.0f) + log1pf(expf(-fabsf(x)));
}
__device__ __forceinline__ float silu_f(float x) {
  return x / (1.0f + expf(-x));
}

// ---------------------------------------------------------------------------
// WMMA GEMM: Y[M,N] = A[M,K] * W[N,K]^T  (+ fused epilogue), fp32 matrix pipe.
// One wave owns a 16 x (16*NT) strip: one A fragment per K-step feeds NT
// back-to-back V_WMMA_F32_16X16X4_F32 ops. K is a compile-time constant so
// unrolled loads use immediate offsets; all addressing is integer offsets off
// the __restrict__ kernel args so loads lower to global_load_b64 (not flat —
// flat would tie LOADcnt to DScnt and the LDS port, ISA §10.1.1).
// Requirements: M % 16 == 0, K % 4 == 0 (true for every call site).
// Ragged N handled by clamping the W row index (loads stay unconditional) and
// guarding only the stores — no predicated loads, no exec-mask churn.
// Fragment layouts per CDNA5 ISA 7.12.2:
//   A 16x4 : lanes 0-15 hold M=lane, {K=0,1}; lanes 16-31 hold {K=2,3}
//   B 4x16 : lanes 0-15 hold N=lane, rows {K=0,1}; lanes 16-31 rows {K=2,3}
//   C/D    : vgpr v = row m0 + (lane>=16)*8 + v, col n0 + (lane&15)
// AT=true reads A as A[k*lda + m] (channel-major input, embed stage).
// ---------------------------------------------------------------------------
template <bool AT, int NT, int K, int FLAGS>
__global__ __launch_bounds__(32) void gemm_wmma(
    const float* __restrict__ A, long lda, long sAz,
    const float* __restrict__ Wt, long ldw,
    float* __restrict__ Y, long ldy, long sYz,
    const float* __restrict__ bias,
    const float* __restrict__ Res, long ldr,
    int M, int N) {
  const int t = threadIdx.x;
  const int half = t >> 4;
  const int r = t & 15;
  const int m0 = blockIdx.y * 16;
  const int n0 = blockIdx.x * (16 * NT);
  const int mrow = m0 + r;  // always < M (M % 16 == 0)

  // Per-lane integer offsets (GEPs stay rooted at the kernel args).
  long aoff;
  if (AT) {
    aoff = (long)blockIdx.z * sAz + (long)(2 * half) * lda + mrow;
  } else {
    aoff = (long)blockIdx.z * sAz + (long)mrow * lda + 2 * half;
  }
  long woff[NT];
#pragma unroll
  for (int j = 0; j < NT; ++j) {
    int nc = n0 + j * 16 + r;
    nc = nc < N ? nc : (N - 1);  // clamp: loads stay unconditional
    woff[j] = (long)nc * ldw + 2 * half;
  }

  v8f acc[NT];
#pragma unroll
  for (int j = 0; j < NT; ++j) acc[j] = v8f{};

#pragma unroll 4
  for (int k0 = 0; k0 < K; k0 += 4) {
    v2f a;
    if (AT) {
      a[0] = A[aoff];
      a[1] = A[aoff + lda];
      aoff += 4 * lda;
    } else {
      a = *(const v2f*)(A + aoff + k0);
    }
    v2f b[NT];
#pragma unroll
    for (int j = 0; j < NT; ++j) b[j] = *(const v2f*)(Wt + woff[j] + k0);
#pragma unroll
    for (int j = 0; j < NT; ++j) {
      // 8 args: (neg_a, A, neg_b, B, c_mod, C, reuse_a, reuse_b)
      acc[j] = __builtin_amdgcn_wmma_f32_16x16x4_f32(
          false, a, false, b[j], (short)0, acc[j], false, false);
    }
  }

  Y += (long)blockIdx.z * sYz;
#pragma unroll
  for (int j = 0; j < NT; ++j) {
    const int col = n0 + j * 16 + r;
    if (col >= N) continue;
    const float bv = (FLAGS & GF_BIAS) ? bias[col] : 0.0f;
#pragma unroll
    for (int v = 0; v < 8; ++v) {
      const int row = m0 + half * 8 + v;
      float val = acc[j][v] + bv;
      if (FLAGS & GF_GELU) val = gelu_exact(val);
      if (FLAGS & GF_SOFTPLUS) val = softplus_f(val);
      if (FLAGS & GF_RES) val += Res[(long)row * ldr + col];
      if (FLAGS & GF_ACC) val += Y[(long)row * ldy + col];
      Y[(long)row * ldy + col] = val;
    }
  }
}

// ---------------------------------------------------------------------------
// LayerNorm over C=128, one wave32 per token, shuffle-tree reductions.
// ---------------------------------------------------------------------------
__global__ __launch_bounds__(256) void ln_kernel(
    const float* __restrict__ X, const float* __restrict__ g,
    const float* __restrict__ b, float* __restrict__ Y, int ntok) {
  const int wave = threadIdx.x >> 5;
  const int lane = threadIdx.x & 31;
  const int tok = blockIdx.x * (blockDim.x >> 5) + wave;
  if (tok >= ntok) return;
  const float* xr = X + (unsigned)tok * Cn;
  float vals[4];
  float s = 0.0f;
#pragma unroll
  for (int i = 0; i < 4; ++i) {
    vals[i] = xr[lane + 32 * i];
    s += vals[i];
  }
#pragma unroll
  for (int o = 16; o > 0; o >>= 1) s += __shfl_xor(s, o, 32);
  const float mean = s * (1.0f / Cn);
  float vs = 0.0f;
#pragma unroll
  for (int i = 0; i < 4; ++i) {
    const float d = vals[i] - mean;
    vs += d * d;
  }
#pragma unroll
  for (int o = 16; o > 0; o >>= 1) vs += __shfl_xor(vs, o, 32);
  const float inv = rsqrtf(vs * (1.0f / Cn) + 1e-5f);
  float* yr = Y + (unsigned)tok * Cn;
#pragma unroll
  for (int i = 0; i < 4; ++i) {
    const int c = lane + 32 * i;
    yr[c] = (vals[i] - mean) * inv * g[c] + b[c];
  }
}

// ---------------------------------------------------------------------------
// Depthwise causal conv (DK=4) + SiLU. dir=0: taps x[l-3..l]; dir=1 is the
// flip-conv-flip equivalent: taps x[l..l+3].
// ---------------------------------------------------------------------------
__global__ __launch_bounds__(256) void conv_silu_kernel(
    const float* __restrict__ xz, const float* __restrict__ cw,
    const float* __restrict__ cb, float* __restrict__ xc, int dir) {
  const unsigned idx = blockIdx.x * 256u + threadIdx.x;
  if (idx >= (unsigned)BL * DIn) return;
  const int d = (int)(idx % DIn);
  const int l = (int)((idx / DIn) % Ln);
  const int b = (int)(idx / ((unsigned)DIn * Ln));
  float acc = cb[d];
#pragma unroll
  for (int k = 0; k < DKn; ++k) {
    const int p = dir ? (l + (DKn - 1) - k) : (l - (DKn - 1) + k);
    if (p >= 0 && p < Ln)
      acc += cw[d * DKn + k] * xz[(unsigned)(b * Ln + p) * (2 * DIn) + d];
  }
  xc[idx] = silu_f(acc);
}

// ---------------------------------------------------------------------------
// Selective scan: per (b,d) thread keeps h[S=16] in registers, walks L=4096
// steps in the chosen direction. y = (sum_s h*C + Dp*xc) * silu(z).
// States are independent across d, so we split DI over blocks for occupancy.
// ---------------------------------------------------------------------------
__global__ __launch_bounds__(64) void scan_kernel(
    const float* __restrict__ dt, const float* __restrict__ dbc,
    const float* __restrict__ xc, const float* __restrict__ xz,
    const float* __restrict__ Alog, const float* __restrict__ Dp,
    float* __restrict__ y, int dir) {
  const int chunks = DIn / 64;
  const int b = blockIdx.x / chunks;
  const int d = (blockIdx.x % chunks) * 64 + threadIdx.x;
  float A[Sn];
#pragma unroll
  for (int s = 0; s < Sn; ++s) A[s] = -expf(Alog[(unsigned)d * Sn + s]);
  float h[Sn];
#pragma unroll
  for (int s = 0; s < Sn; ++s) h[s] = 0.0f;
  const float dp = Dp[d];
  for (int i = 0; i < Ln; ++i) {
    const int l = dir ? (Ln - 1 - i) : i;
    const unsigned tix = (unsigned)b * Ln + l;
    const float dtv = dt[tix * DIn + d];
    const float xcv = xc[tix * DIn + d];
    const float zv = xz[tix * (2 * DIn) + DIn + d];
    const float* Bm = dbc + tix * XPn + Rn;
    const float* Cm = Bm + Sn;
    float accv = 0.0f;
#pragma unroll
    for (int s = 0; s < Sn; ++s) {
      const float dA = expf(dtv * A[s]);
      h[s] = dA * h[s] + dtv * Bm[s] * xcv;
      accv += h[s] * Cm[s];
    }
    y[tix * DIn + d] = (accv + dp * xcv) * silu_f(zv);
  }
}

// ---------------------------------------------------------------------------
// Final fuse: out[b,c,l] = x[b,c,l] + sigmoid(gate[b,l,c]) * enh[b,l,c]
// (also performs the (B,L,C) -> (B,C,H,W) transpose).
// ---------------------------------------------------------------------------
__global__ __launch_bounds__(256) void final_kernel(
    const float* __restrict__ xin, const float* __restrict__ gate,
    const float* __restrict__ enh, float* __restrict__ out) {
  const unsigned idx = blockIdx.x * 256u + threadIdx.x;
  if (idx >= (unsigned)Bn * Cn * Ln) return;
  const int l = (int)(idx % Ln);
  const int c = (int)((idx / Ln) % Cn);
  const int b = (int)(idx / ((unsigned)Ln * Cn));
  const unsigned tix = ((unsigned)b * Ln + l) * Cn + c;
  const float g = 1.0f / (1.0f + expf(-gate[tix]));
  out[idx] = xin[idx] + g * enh[tix];
}

inline dim3 gemm_grid(int M, int N, int NT, int Z) {
  return dim3((N + 16 * NT - 1) / (16 * NT), M / 16, Z);
}

}  // namespace

extern "C" void kernel_launch(void* const* d_in, const int* in_sizes, int n_in,
                              void* d_out, int out_size, void* d_ws,
                              size_t ws_size, hipStream_t stream) {
  const float* x        = (const float*)d_in[0];
  const float* embed_w  = (const float*)d_in[1];
  const float* embed_b  = (const float*)d_in[2];
  const float* outp_w   = (const float*)d_in[3];
  const float* outp_b   = (const float*)d_in[4];
  const float* gate_w   = (const float*)d_in[5];
  const float* gate_b   = (const float*)d_in[6];
  const float* norm_g   = (const float*)d_in[7];
  const float* norm_b   = (const float*)d_in[8];
  const float* in_proj  = (const float*)d_in[9];
  const float* conv_w   = (const float*)d_in[10];
  const float* conv_b   = (const float*)d_in[11];
  const float* x_proj   = (const float*)d_in[12];
  const float* dt_w     = (const float*)d_in[13];
  const float* dt_b     = (const float*)d_in[14];
  const float* A_log    = (const float*)d_in[15];
  const float* Dp       = (const float*)d_in[16];
  const float* out_w    = (const float*)d_in[17];
  const float* ffn_w1   = (const float*)d_in[18];
  const float* ffn_b1   = (const float*)d_in[19];
  const float* ffn_w2   = (const float*)d_in[20];
  const float* ffn_b2   = (const float*)d_in[21];
  float* out = (float*)d_out;

  // Workspace carve-up (floats). Everything stays resident in L2 (192 MB).
  float* p = (float*)d_ws;
  float* xe   = p; p += (long)BL * Cn;        // embed output (kept for gate)
  float* bufA = p; p += (long)BL * Cn;        // layer ping
  float* bufB = p; p += (long)BL * Cn;        // layer pong
  float* hbuf = p; p += (long)BL * Cn;        // LN output
  float* xb2  = p; p += (long)BL * Cn;        // post-mamba residual stream
  float* xzb  = p; p += (long)BL * 2 * DIn;   // in_proj output [xin | z]
  float* xcb  = p; p += (long)BL * DIn;       // conv+silu output
  float* dbc  = p; p += (long)BL * XPn;       // x_proj output [dt8|B16|C16]
  float* dtb  = p; p += (long)BL * DIn;       // softplus(dt)
  float* yss  = p; p += (long)BL * DIn;       // scan output
  float* ffn1 = p; p += (long)BL * FFn;       // FFN hidden
  float* enh  = p; p += (long)BL * Cn;        // output projection
  float* gat  = p; p += (long)BL * Cn;        // gate logits
  (void)ws_size; (void)n_in; (void)in_sizes; (void)out_size;

  const dim3 wave(32);

  // 1) Embed: xe[b,l,o] = sum_c x[b,c,l]*embed_w[o,c] + embed_b  (A transposed)
  gemm_wmma<true, 4, Cn, GF_BIAS>
      <<<gemm_grid(Ln, Cn, 4, Bn), wave, 0, stream>>>(
          x, Ln, (long)Cn * Ln, embed_w, Cn, xe, Cn, (long)Ln * Cn, embed_b,
          nullptr, 0, Ln, Cn);

  const float* cur = xe;  // current residual stream (B,L,C)
  float* nxt = bufA;

  for (int i = 0; i < NLn; ++i) {
    const float* ng  = norm_g + (long)i * Cn;
    const float* nb  = norm_b + (long)i * Cn;
    const float* wip = in_proj + (long)i * 2 * DIn * Cn;
    const float* wc  = conv_w + (long)i * DIn * DKn;
    const float* bc  = conv_b + (long)i * DIn;
    const float* wxp = x_proj + (long)i * XPn * DIn;
    const float* wdt = dt_w + (long)i * DIn * Rn;
    const float* bdt = dt_b + (long)i * DIn;
    const float* al  = A_log + (long)i * DIn * Sn;
    const float* dp  = Dp + (long)i * DIn;
    const float* wmo = out_w + (long)i * Cn * DIn;
    const float* w1  = ffn_w1 + (long)i * FFn * Cn;
    const float* b1  = ffn_b1 + (long)i * FFn;
    const float* w2  = ffn_w2 + (long)i * Cn * FFn;
    const float* b2  = ffn_b2 + (long)i * Cn;

    for (int dir = 0; dir < 2; ++dir) {
      // LN1
      ln_kernel<<<(BL + 7) / 8, 256, 0, stream>>>(cur, ng, nb, hbuf, BL);
      // in_proj: (BL,128) x (512,128)^T -> xz (BL,512)
      gemm_wmma<false, 4, Cn, 0>
          <<<gemm_grid(BL, 2 * DIn, 4, 1), wave, 0, stream>>>(
              hbuf, Cn, 0, wip, Cn, xzb, 2 * DIn, 0, nullptr, nullptr, 0,
              BL, 2 * DIn);
      // depthwise conv + SiLU (direction-aware)
      conv_silu_kernel<<<(BL * DIn + 255) / 256, 256, 0, stream>>>(
          xzb, wc, bc, xcb, dir);
      // x_proj: (BL,256) x (40,256)^T -> dbc (BL,40)
      gemm_wmma<false, 1, DIn, 0>
          <<<gemm_grid(BL, XPn, 1, 1), wave, 0, stream>>>(
              xcb, DIn, 0, wxp, DIn, dbc, XPn, 0, nullptr, nullptr, 0,
              BL, XPn);
      // dt: softplus((BL,8) x (256,8)^T + dt_b) -> (BL,256)
      gemm_wmma<false, 4, Rn, GF_BIAS | GF_SOFTPLUS>
          <<<gemm_grid(BL, DIn, 4, 1), wave, 0, stream>>>(
              dbc, XPn, 0, wdt, Rn, dtb, DIn, 0, bdt, nullptr, 0,
              BL, DIn);
      // selective scan (direction-aware), gated by silu(z)
      scan_kernel<<<Bn * (DIn / 64), 64, 0, stream>>>(
          dtb, dbc, xcb, xzb, al, dp, yss, dir);
      // mamba out proj + residual: xb2 = y * W^T + cur
      gemm_wmma<false, 4, DIn, GF_RES>
          <<<gemm_grid(BL, Cn, 4, 1), wave, 0, stream>>>(
              yss, DIn, 0, wmo, DIn, xb2, Cn, 0, nullptr, cur, Cn,
              BL, Cn);
      // LN2
      ln_kernel<<<(BL + 7) / 8, 256, 0, stream>>>(xb2, ng, nb, hbuf, BL);
      // FFN1 + exact GELU
      gemm_wmma<false, 4, Cn, GF_BIAS | GF_GELU>
          <<<gemm_grid(BL, FFn, 4, 1), wave, 0, stream>>>(
              hbuf, Cn, 0, w1, Cn, ffn1, FFn, 0, b1, nullptr, 0,
              BL, FFn);
      // FFN2 + bias + residual(xb2); dir==1 accumulates onto dir==0's output,
      // realizing blk(x) + flip(blk(flip(x))) without materializing flips.
      if (dir == 0) {
        gemm_wmma<false, 4, FFn, GF_BIAS | GF_RES>
            <<<gemm_grid(BL, Cn, 4, 1), wave, 0, stream>>>(
                ffn1, FFn, 0, w2, FFn, nxt, Cn, 0, b2, xb2, Cn,
                BL, Cn);
      } else {
        gemm_wmma<false, 4, FFn, GF_BIAS | GF_RES | GF_ACC>
            <<<gemm_grid(BL, Cn, 4, 1), wave, 0, stream>>>(
                ffn1, FFn, 0, w2, FFn, nxt, Cn, 0, b2, xb2, Cn,
                BL, Cn);
      }
    }
    cur = nxt;
    nxt = (nxt == bufA) ? bufB : bufA;
  }

  // Output projection and gate (both read-only on preserved buffers)
  gemm_wmma<false, 4, Cn, GF_BIAS>
      <<<gemm_grid(BL, Cn, 4, 1), wave, 0, stream>>>(
          cur, Cn, 0, outp_w, Cn, enh, Cn, 0, outp_b, nullptr, 0, BL, Cn);
  gemm_wmma<false, 4, Cn, GF_BIAS>
      <<<gemm_grid(BL, Cn, 4, 1), wave, 0, stream>>>(
          xe, Cn, 0, gate_w, Cn, gat, Cn, 0, gate_b, nullptr, 0, BL, Cn);

  final_kernel<<<(Bn * Cn * Ln + 255) / 256, 256, 0, stream>>>(
      x, gat, enh, out);
}